// AGNNConv_68659347194082
// MI455X (gfx1250) — compile-verified
//
#include <hip/hip_runtime.h>
#include <math.h>

typedef __attribute__((ext_vector_type(2))) float v2f;
typedef __attribute__((ext_vector_type(8))) float v8f;

#define FEAT_D 64
#define AGNN_EPS 1e-12f

// Order-preserving float <-> uint mapping for atomicMax-based segment max.
__device__ __forceinline__ unsigned orderF(float x) {
    unsigned b = __float_as_uint(x);
    return (b & 0x80000000u) ? ~b : (b | 0x80000000u);
}
__device__ __forceinline__ float unorderF(unsigned b) {
    return (b & 0x80000000u) ? __uint_as_float(b & 0x7FFFFFFFu)
                             : __uint_as_float(~b);
}

// ---------------------------------------------------------------------------
// Kernel 1: L2-normalize rows (one wave32 per row), zero d_out, init emax/psum
// ---------------------------------------------------------------------------
__global__ void k_norm_init(const float* __restrict__ feat,
                            float* __restrict__ nh,
                            float* __restrict__ out,
                            unsigned* __restrict__ emax,
                            float* __restrict__ psum, int N) {
    int wid  = (int)((blockIdx.x * blockDim.x + threadIdx.x) >> 5);
    int lane = (int)(threadIdx.x & 31);
    if (wid >= N) return;
    const float2* row = (const float2*)(feat + (size_t)wid * FEAT_D);
    float2 v = row[lane];
    float ss = v.x * v.x + v.y * v.y;
    #pragma unroll
    for (int off = 16; off > 0; off >>= 1) ss += __shfl_xor(ss, off, 32);
    float nrm   = sqrtf(ss);
    float scale = 1.0f / fmaxf(nrm, AGNN_EPS);
    float2 o; o.x = v.x * scale; o.y = v.y * scale;
    ((float2*)(nh + (size_t)wid * FEAT_D))[lane] = o;
    float2 z; z.x = 0.0f; z.y = 0.0f;
    ((float2*)(out + (size_t)wid * FEAT_D))[lane] = z;
    if (lane == 0) { emax[wid] = 0u; psum[wid] = 0.0f; }
}

// ---------------------------------------------------------------------------
// Kernel 2: edge scores. One wave per 16 edges via V_WMMA_F32_16X16X4_F32.
// A = 16 src-rows (16xK), B = 16 dst-rows as columns (Kx16); diagonal of the
// 16x16 product = per-edge dots. K looped 64/4 = 16 WMMA steps, f32 exact.
// ---------------------------------------------------------------------------
#if __has_builtin(__builtin_amdgcn_wmma_f32_16x16x4_f32)
#define USE_WMMA_SCORE 1
__global__ void k_score_wmma(const float* __restrict__ nh,
                             const int* __restrict__ src,
                             const int* __restrict__ dst,
                             const float* __restrict__ beta,
                             float* __restrict__ escore,
                             unsigned* __restrict__ emax, int E) {
    int tile = (int)((blockIdx.x * blockDim.x + threadIdx.x) >> 5);
    int lane = (int)(threadIdx.x & 31);
    int base = tile * 16;
    if (base >= E) return;                 // wave-uniform: EXEC stays all-ones
    int m     = lane & 15;                 // edge slot within tile
    int e_idx = base + m;
    int eclmp = min(e_idx, E - 1);         // tail-safe loads, results masked
    int srow  = src[eclmp];
    int drow  = dst[eclmp];
    int kb    = (lane >> 4) << 1;          // 0 for lanes 0-15, 2 for 16-31
    const float* ap = nh + (size_t)srow * FEAT_D + kb;  // A: M=m, K=k0+kb(+1)
    const float* bp = nh + (size_t)drow * FEAT_D + kb;  // B: N=m, K=k0+kb(+1)
    v8f c = {};
    #pragma unroll
    for (int k0 = 0; k0 < FEAT_D; k0 += 4) {
        float2 av = *(const float2*)(ap + k0);
        float2 bv = *(const float2*)(bp + k0);
        v2f a; a[0] = av.x; a[1] = av.y;
        v2f b; b[0] = bv.x; b[1] = bv.y;
        c = __builtin_amdgcn_wmma_f32_16x16x4_f32(false, a, false, b,
                                                  (short)0, c, false, false);
    }
    // Diagonal extract: diag[m] lives at (VGPR m&7, lane m) for m<8,
    // (VGPR m&7, lane m+16) for m>=8.
    int r = lane & 7;
    float vsel = c[0];
    vsel = (r == 1) ? c[1] : vsel;
    vsel = (r == 2) ? c[2] : vsel;
    vsel = (r == 3) ? c[3] : vsel;
    vsel = (r == 4) ? c[4] : vsel;
    vsel = (r == 5) ? c[5] : vsel;
    vsel = (r == 6) ? c[6] : vsel;
    vsel = (r == 7) ? c[7] : vsel;
    int srcLane = (lane < 8) ? lane : lane + 16;
    float diag = __shfl(vsel, srcLane, 32);
    if (lane < 16 && e_idx < E) {
        float ev = beta[0] * diag;
        escore[e_idx] = ev;
        atomicMax(emax + drow, orderF(ev));
    }
}
#else
#define USE_WMMA_SCORE 0
// Fallback: one wave per edge, VALU dot + butterfly reduce.
__global__ void k_score_dot(const float* __restrict__ nh,
                            const int* __restrict__ src,
                            const int* __restrict__ dst,
                            const float* __restrict__ beta,
                            float* __restrict__ escore,
                            unsigned* __restrict__ emax, int E) {
    int e    = (int)((blockIdx.x * blockDim.x + threadIdx.x) >> 5);
    int lane = (int)(threadIdx.x & 31);
    if (e >= E) return;
    int s = src[e], d = dst[e];
    float2 a = ((const float2*)(nh + (size_t)s * FEAT_D))[lane];
    float2 b = ((const float2*)(nh + (size_t)d * FEAT_D))[lane];
    float ss = a.x * b.x + a.y * b.y;
    #pragma unroll
    for (int off = 16; off > 0; off >>= 1) ss += __shfl_xor(ss, off, 32);
    if (lane == 0) {
        float ev = beta[0] * ss;
        escore[e] = ev;
        atomicMax(emax + d, orderF(ev));
    }
}
#endif

// ---------------------------------------------------------------------------
// Kernel 3: p = exp(e - emax[dst]); psum[dst] += p  (one thread per edge)
// ---------------------------------------------------------------------------
__global__ void k_exp_sum(const int* __restrict__ dst,
                          float* __restrict__ escore,
                          const unsigned* __restrict__ emax,
                          float* __restrict__ psum, int E) {
    int e = (int)(blockIdx.x * blockDim.x + threadIdx.x);
    if (e >= E) return;
    int d = dst[e];
    float mx = unorderF(emax[d]);   // node has >=1 in-edge here, so finite
    float p  = expf(escore[e] - mx);
    escore[e] = p;
    atomicAdd(psum + d, p);
}

// ---------------------------------------------------------------------------
// Kernel 4: out[dst] += (p/psum[dst]) * feat[src]  (one wave per edge,
// coalesced 256B row gather, 2 f32 atomics per lane into L2)
// ---------------------------------------------------------------------------
__global__ void k_aggregate(const float* __restrict__ feat,
                            const int* __restrict__ src,
                            const int* __restrict__ dst,
                            const float* __restrict__ pbuf,
                            const float* __restrict__ psum,
                            float* __restrict__ out, int E) {
    int e    = (int)((blockIdx.x * blockDim.x + threadIdx.x) >> 5);
    int lane = (int)(threadIdx.x & 31);
    if (e >= E) return;
    int s = src[e], d = dst[e];
    float w = pbuf[e] / fmaxf(psum[d], AGNN_EPS);
    float2 f = ((const float2*)(feat + (size_t)s * FEAT_D))[lane];
    float* op = out + (size_t)d * FEAT_D + lane * 2;
    atomicAdd(op,     w * f.x);
    atomicAdd(op + 1, w * f.y);
}

// ---------------------------------------------------------------------------
extern "C" void kernel_launch(void* const* d_in, const int* in_sizes, int n_in,
                              void* d_out, int out_size, void* d_ws, size_t ws_size,
                              hipStream_t stream) {
    const float* feat = (const float*)d_in[0];
    const float* beta = (const float*)d_in[1];
    const int*   src  = (const int*)d_in[2];
    const int*   dst  = (const int*)d_in[3];
    const int N = in_sizes[0] / FEAT_D;
    const int E = in_sizes[2];
    float* out = (float*)d_out;

    // workspace carve-out: nh[N*64] | escore[E] | emax[N] | psum[N]
    char* ws = (char*)d_ws;
    size_t off = 0;
    float* nh = (float*)(ws + off);      off += (size_t)N * FEAT_D * sizeof(float);
    float* escore = (float*)(ws + off);  off += (size_t)E * sizeof(float);
    off = (off + 255) & ~(size_t)255;
    unsigned* emax = (unsigned*)(ws + off); off += (size_t)N * sizeof(unsigned);
    float* psum = (float*)(ws + off);

    const int TPB = 256;            // 8 wave32 per block
    const int WPB = TPB / 32;

    int gNorm = (N + WPB - 1) / WPB;
    k_norm_init<<<gNorm, TPB, 0, stream>>>(feat, nh, out, emax, psum, N);

#if USE_WMMA_SCORE
    int tiles  = (E + 15) / 16;
    int gScore = (tiles + WPB - 1) / WPB;
    k_score_wmma<<<gScore, TPB, 0, stream>>>(nh, src, dst, beta, escore, emax, E);
#else
    int gScore = (E + WPB - 1) / WPB;
    k_score_dot<<<gScore, TPB, 0, stream>>>(nh, src, dst, beta, escore, emax, E);
#endif

    k_exp_sum<<<(E + TPB - 1) / TPB, TPB, 0, stream>>>(dst, escore, emax, psum, E);

    k_aggregate<<<(E + WPB - 1) / WPB, TPB, 0, stream>>>(feat, src, dst,
                                                         escore, psum, out, E);
}